// Decoder_15109694948046
// MI455X (gfx1250) — compile-verified
//
#include <hip/hip_runtime.h>
#include <hip/hip_bf16.h>
#include <math.h>

#define BB 16
#define HH 256
#define EE 128
#define VV 50000
#define NOOVN 32
#define SECLN 8
#define WORDLN 512
#define SSN (SECLN*WORDLN)        /* 4096 */
#define D2H 512                   /* 2H  */
#define KCAT 768                  /* 3H  */
#define VEXT (VV+NOOVN)           /* 50032 */
#define EOS_TOK 3
#define FSZ 2

/* d_out layout (floats): final, h_new, c_new, context, attn, coverage, focus, sad */
#define OUT_FINAL 0
#define OUT_HNEW  (BB*VEXT)
#define OUT_CNEW  (OUT_HNEW + BB*HH)
#define OUT_CTX   (OUT_CNEW + BB*HH)
#define OUT_ATTN  (OUT_CTX + BB*D2H)
#define OUT_COV   (OUT_ATTN + BB*SSN)
#define OUT_FOCUS (OUT_COV + BB*SSN)
#define OUT_SAD   (OUT_FOCUS + BB*SECLN)

typedef float v2f __attribute__((ext_vector_type(2)));
typedef float v8f __attribute__((ext_vector_type(8)));

/* ---------------- 1: x = [prev_context, emb] @ W_comb^T + b_comb ---------------- */
__global__ __launch_bounds__(128) void k_x(const float* __restrict__ prev_context,
                                           const int* __restrict__ inp,
                                           const float* __restrict__ emb_table,
                                           const float* __restrict__ W_comb,
                                           const float* __restrict__ b_comb,
                                           float* __restrict__ x) {
  int b = blockIdx.x, j = threadIdx.x;
  const float* w  = W_comb + j * (D2H + EE);
  const float* pc = prev_context + b * D2H;
  const float* e  = emb_table + (size_t)inp[b] * EE;
  float acc = b_comb[j];
  for (int k = 0; k < D2H; ++k) acc += pc[k] * w[k];
  for (int k = 0; k < EE;  ++k) acc += e[k]  * w[D2H + k];
  x[b * EE + j] = acc;
}

/* ---------------- 2: LSTM step, dec_state, dec_cat[:,0:256] ---------------- */
__global__ __launch_bounds__(256) void k_lstm(const float* __restrict__ x,
                                              const float* __restrict__ h,
                                              const float* __restrict__ c,
                                              const float* __restrict__ W_ih,
                                              const float* __restrict__ W_hh,
                                              const float* __restrict__ b_ih,
                                              const float* __restrict__ b_hh,
                                              float* __restrict__ h_new,
                                              float* __restrict__ c_new,
                                              float* __restrict__ dec_state,
                                              float* __restrict__ dec_cat) {
  int b = blockIdx.x, t = threadIdx.x;
  const float* xb = x + b * EE;
  const float* hb = h + b * HH;
  float g[4];
#pragma unroll
  for (int gi = 0; gi < 4; ++gi) {
    int row = gi * HH + t;
    const float* wi = W_ih + (size_t)row * EE;
    const float* wh = W_hh + (size_t)row * HH;
    float acc = b_ih[row] + b_hh[row];
    for (int k = 0; k < EE; ++k) acc += xb[k] * wi[k];
    for (int k = 0; k < HH; ++k) acc += hb[k] * wh[k];
    g[gi] = acc;
  }
  float ig = 1.f / (1.f + __expf(-g[0]));
  float fg = 1.f / (1.f + __expf(-g[1]));
  float gg = tanhf(g[2]);
  float og = 1.f / (1.f + __expf(-g[3]));
  float cn = fg * c[b * HH + t] + ig * gg;
  float hn = og * tanhf(cn);
  h_new[b * HH + t] = hn;
  c_new[b * HH + t] = cn;
  dec_state[b * D2H + t] = hn;
  dec_state[b * D2H + HH + t] = cn;
  dec_cat[b * KCAT + t] = hn;
}

/* ---------------- 3: section attention scores sc[b,s] ---------------- */
__global__ __launch_bounds__(256) void k_sec(const float* __restrict__ enc_sec,
                                             const float* __restrict__ W_feat,
                                             const float* __restrict__ dec_state,
                                             const float* __restrict__ W_d,
                                             const float* __restrict__ b_d,
                                             const float* __restrict__ v_sec,
                                             float* __restrict__ sc) {
  int b = blockIdx.x >> 3, s = blockIdx.x & 7, t = threadIdx.x;
  const float* es = enc_sec + (size_t)(b * SECLN + s) * D2H;
  const float* ds = dec_state + b * D2H;
  float part = 0.f;
#pragma unroll
  for (int jj = 0; jj < 2; ++jj) {
    int j = t + jj * 256;
    const float* wf = W_feat + (size_t)j * D2H;
    const float* wd = W_d    + (size_t)j * D2H;
    float acc = b_d[j];
    for (int k = 0; k < D2H; ++k) acc += es[k] * wf[k] + ds[k] * wd[k];
    part += tanhf(acc) * v_sec[j];
  }
  __shared__ float red[256];
  red[t] = part; __syncthreads();
  for (int off = 128; off > 0; off >>= 1) {
    if (t < off) red[t] += red[t + off];
    __syncthreads();
  }
  if (t == 0) sc[b * SECLN + s] = red[0];
}

/* ---------------- 4: sad softmax, hard-focus update, top-2 (serial/batch) ---------------- */
__global__ __launch_bounds__(32) void k_focus(const float* __restrict__ sc,
                                              const float* __restrict__ sec_mask,
                                              const float* __restrict__ focus_in,
                                              const int* __restrict__ inp,
                                              float* __restrict__ sad_out,
                                              float* __restrict__ focus_out,
                                              float* __restrict__ topk,
                                              int* __restrict__ gidx) {
  int b = threadIdx.x;
  if (b >= BB) return;
  float sad[SECLN], f[SECLN];
  float mx = -1e30f;
  for (int i = 0; i < SECLN; ++i) { sad[i] = sc[b * SECLN + i]; mx = fmaxf(mx, sad[i]); }
  float sum = 0.f;
  for (int i = 0; i < SECLN; ++i) { sad[i] = __expf(sad[i] - mx); sum += sad[i]; }
  for (int i = 0; i < SECLN; ++i) sad[i] = sad[i] / sum * sec_mask[b * SECLN + i];
  sum = 0.f;
  for (int i = 0; i < SECLN; ++i) sum += sad[i];
  for (int i = 0; i < SECLN; ++i) { sad[i] /= sum; sad_out[b * SECLN + i] = sad[i]; }

  for (int i = 0; i < SECLN; ++i) f[i] = focus_in[b * SECLN + i];
  /* first nonzero of focus */
  int fidx = SECLN;
  for (int i = 0; i < SECLN; ++i) if (f[i] != 0.f) { fidx = i; break; }
  float focus_beta_max = (fidx < SECLN) ? sad[fidx] : 0.f;
  /* rem_beta = sad * (i > fidx); max + first argmax */
  float mrb = 0.f; int mri = 0; {
    float best = (0 > fidx) ? sad[0] : 0.f; int bi = 0;
    for (int i = 1; i < SECLN; ++i) {
      float v = (i > fidx) ? sad[i] : 0.f;
      if (v > best) { best = v; bi = i; }
    }
    mrb = best; mri = bi;
  }
  int compare = (focus_beta_max * 2.f > mrb) ? 1 : 0;
  int src = (mrb != 0.f) ? 1 : 0;
  /* amb = sad * (i > mri, only if src); max + first argmax */
  float amx = 0.f; int ami = 0; {
    float best = 0.f; int bi = 0; int first = 1;
    for (int i = 0; i < SECLN; ++i) {
      float v = (src && i > mri) ? sad[i] : 0.f;
      if (first) { best = v; bi = i; first = 0; }
      else if (v > best) { best = v; bi = i; }
    }
    amx = best; ami = bi;
  }
  int mam = (amx != 0.f) ? 1 : 0;
  float fu[SECLN];
  for (int i = 0; i < SECLN; ++i) fu[i] = 0.f;
  if (compare) {
    if (fidx < SECLN) fu[fidx] += sad[fidx];
    if (src) fu[mri] += sad[mri];
  } else {
    if (src) fu[mri] += sad[mri];
    if (mam) fu[ami] += sad[ami];
  }
  float fs = 0.f;
  for (int i = 0; i < SECLN; ++i) fs += fu[i];
  for (int i = 0; i < SECLN; ++i) fu[i] /= fs;

  int eos = (inp[b] == EOS_TOK);
  float fn[SECLN];
  for (int i = 0; i < SECLN; ++i) {
    fn[i] = eos ? fu[i] : f[i];
    focus_out[b * SECLN + i] = fn[i];
  }
  /* top-2 (stable: first occurrence wins ties) */
  int i0 = 0; float v0 = fn[0];
  for (int i = 1; i < SECLN; ++i) if (fn[i] > v0) { v0 = fn[i]; i0 = i; }
  int i1 = -1; float v1 = -1e30f;
  for (int i = 0; i < SECLN; ++i) if (i != i0 && fn[i] > v1) { v1 = fn[i]; i1 = i; }
  topk[b * FSZ + 0] = v0; topk[b * FSZ + 1] = v1;
  gidx[b * FSZ + 0] = i0; gidx[b * FSZ + 1] = i1;
}

/* ---------------- 5: dfeat = dec_state @ W_dec^T + b_dec ---------------- */
__global__ __launch_bounds__(256) void k_dfeat(const float* __restrict__ dec_state,
                                               const float* __restrict__ W_dec,
                                               const float* __restrict__ b_dec,
                                               float* __restrict__ dfeat) {
  int b = blockIdx.x >> 1;
  int j = (blockIdx.x & 1) * 256 + threadIdx.x;
  const float* ds = dec_state + b * D2H;
  const float* w  = W_dec + (size_t)j * D2H;
  float acc = b_dec[j];
  for (int k = 0; k < D2H; ++k) acc += ds[k] * w[k];
  dfeat[b * D2H + j] = acc;
}

/* ---------------- 6: word scores only for focused sections (wave/word) ---------------- */
__global__ __launch_bounds__(256) void k_wscore(const float* __restrict__ enc_feature,
                                                const float* __restrict__ dfeat,
                                                const float* __restrict__ coverage,
                                                const float* __restrict__ w_cov,
                                                const float* __restrict__ v_att,
                                                const int* __restrict__ gidx,
                                                float* __restrict__ score_g) {
  int wid  = blockIdx.x * 8 + (threadIdx.x >> 5);  /* 0..16383 */
  int lane = threadIdx.x & 31;
  int b = wid >> 10, rem = wid & 1023, fsec = rem >> 9, w = rem & (WORDLN - 1);
  int s = gidx[b * FSZ + fsec] * WORDLN + w;
  const float* ef = enc_feature + ((size_t)b * SSN + s) * D2H;
  const float* df = dfeat + b * D2H;
  float cov = coverage[b * SSN + s];
  float acc = 0.f;
  for (int k = lane; k < D2H; k += 32)
    acc += tanhf(ef[k] + df[k] + cov * w_cov[k]) * v_att[k];
#pragma unroll
  for (int off = 16; off > 0; off >>= 1) acc += __shfl_xor(acc, off, 32);
  if (lane == 0) score_g[b * (FSZ * WORDLN) + rem] = acc;
}

/* ---------------- 7: joint softmax over 1024 focused words -> attn, coverage ---------------- */
__global__ __launch_bounds__(256) void k_wsoftmax(const float* __restrict__ score_g,
                                                  const float* __restrict__ enc_mask,
                                                  const float* __restrict__ coverage,
                                                  const float* __restrict__ topk,
                                                  const int* __restrict__ gidx,
                                                  float* __restrict__ attn_out,
                                                  float* __restrict__ cov_out) {
  int b = blockIdx.x, t = threadIdx.x;
  const int NW = FSZ * WORDLN;  /* 1024 */
  __shared__ float red[256];
  __shared__ float smx, sden, sasum;

  /* pass 1: max over raw scores */
  float mx = -1e30f;
  for (int i = t; i < NW; i += 256) mx = fmaxf(mx, score_g[b * NW + i]);
  red[t] = mx; __syncthreads();
  for (int off = 128; off > 0; off >>= 1) {
    if (t < off) red[t] = fmaxf(red[t], red[t + off]);
    __syncthreads();
  }
  if (t == 0) smx = red[0];
  __syncthreads();
  mx = smx;

  /* pass 2: sum of e*m */
  float ps = 0.f;
  for (int i = t; i < NW; i += 256) {
    int fsec = i >> 9, w = i & (WORDLN - 1);
    int s = gidx[b * FSZ + fsec] * WORDLN + w;
    float e = __expf(score_g[b * NW + i] - mx);
    ps += e * enc_mask[b * SSN + s];
  }
  red[t] = ps; __syncthreads();
  for (int off = 128; off > 0; off >>= 1) {
    if (t < off) red[t] += red[t + off];
    __syncthreads();
  }
  if (t == 0) sden = red[0];
  __syncthreads();
  float den = sden;

  /* pass 3: sum of af = topk[f]*q */
  float pa = 0.f;
  for (int i = t; i < NW; i += 256) {
    int fsec = i >> 9, w = i & (WORDLN - 1);
    int s = gidx[b * FSZ + fsec] * WORDLN + w;
    float e = __expf(score_g[b * NW + i] - mx);
    pa += topk[b * FSZ + fsec] * (e * enc_mask[b * SSN + s] / den);
  }
  red[t] = pa; __syncthreads();
  for (int off = 128; off > 0; off >>= 1) {
    if (t < off) red[t] += red[t + off];
    __syncthreads();
  }
  if (t == 0) sasum = red[0];
  __syncthreads();
  float asum = sasum;

  /* zero attn row and copy coverage */
  for (int i = t; i < SSN; i += 256) {
    attn_out[b * SSN + i] = 0.f;
    cov_out[b * SSN + i] = coverage[b * SSN + i];
  }
  __syncthreads();
  /* write focused positions */
  for (int i = t; i < NW; i += 256) {
    int fsec = i >> 9, w = i & (WORDLN - 1);
    int s = gidx[b * FSZ + fsec] * WORDLN + w;
    float e = __expf(score_g[b * NW + i] - mx);
    float a = topk[b * FSZ + fsec] * (e * enc_mask[b * SSN + s] / den) / asum;
    attn_out[b * SSN + s] = a;
    cov_out[b * SSN + s] = coverage[b * SSN + s] + a;
  }
}

/* ---------------- 8: context from focused rows, fill dec_cat[:,256:768] ---------------- */
__global__ __launch_bounds__(512) void k_context(const float* __restrict__ attn_out,
                                                 const float* __restrict__ enc_outputs,
                                                 const int* __restrict__ gidx,
                                                 float* __restrict__ ctx_out,
                                                 float* __restrict__ dec_cat) {
  int b = blockIdx.x, t = threadIdx.x;
  __shared__ float a_sh[FSZ * WORDLN];
  __shared__ int s_sh[FSZ * WORDLN];
  for (int i = t; i < FSZ * WORDLN; i += 512) {
    int fsec = i >> 9, w = i & (WORDLN - 1);
    int s = gidx[b * FSZ + fsec] * WORDLN + w;
    s_sh[i] = s;
    a_sh[i] = attn_out[b * SSN + s];
  }
  __syncthreads();
  float acc = 0.f;
  for (int i = 0; i < FSZ * WORDLN; ++i)
    acc += a_sh[i] * enc_outputs[((size_t)b * SSN + s_sh[i]) * D2H + t];
  ctx_out[b * D2H + t] = acc;
  dec_cat[b * KCAT + HH + t] = acc;
}

/* ---------------- 9: p_gen (wave per batch) ---------------- */
__global__ __launch_bounds__(512) void k_pgen(const float* __restrict__ ctx,
                                              const float* __restrict__ dec_state,
                                              const float* __restrict__ x,
                                              const float* __restrict__ W_pgen,
                                              const float* __restrict__ b_pgen,
                                              float* __restrict__ pgen) {
  int b = threadIdx.x >> 5, lane = threadIdx.x & 31;
  float acc = 0.f;
  for (int k = lane; k < D2H; k += 32) acc += ctx[b * D2H + k] * W_pgen[k];
  for (int k = lane; k < D2H; k += 32) acc += dec_state[b * D2H + k] * W_pgen[D2H + k];
  for (int k = lane; k < EE; k += 32) acc += x[b * EE + k] * W_pgen[2 * D2H + k];
#pragma unroll
  for (int off = 16; off > 0; off >>= 1) acc += __shfl_xor(acc, off, 32);
  if (lane == 0) pgen[b] = 1.f / (1.f + __expf(-(acc + b_pgen[0])));
}

/* ---------------- 10: vocab GEMM via v_wmma_f32_16x16x4_f32 ----------------
 * logits[16,50000] = dec_cat[16,768] @ W_out^T + b_out
 * One 16x16 output tile per wave; A staged in LDS; K=768 -> 192 WMMAs/tile.  */
__global__ __launch_bounds__(256) void k_gemm_wmma(const float* __restrict__ A,     /* dec_cat [16][768] */
                                                   const float* __restrict__ W,     /* W_out [50000][768] */
                                                   const float* __restrict__ bias,  /* b_out [50000] */
                                                   float* __restrict__ logits) {    /* [16][50000] */
  __shared__ float lds[BB * KCAT];  /* 48 KB */
  for (int i = threadIdx.x; i < BB * KCAT; i += 256) lds[i] = A[i];
  __syncthreads();
  int wave = threadIdx.x >> 5;
  int lane = threadIdx.x & 31;
  int tile = blockIdx.x * 8 + wave;
  if (tile >= VV / 16) return;
  int n0 = tile * 16;
  int m = lane & 15;            /* A row (lanes 0-15) / B column, per ISA layout */
  int khalf = (lane >> 4) * 2;  /* lanes 16-31 carry K+2,K+3 */
  const float* wrow = W + (size_t)(n0 + m) * KCAT + khalf;
  const float* arow = lds + m * KCAT + khalf;
  v8f acc = {0.f, 0.f, 0.f, 0.f, 0.f, 0.f, 0.f, 0.f};
#pragma unroll 4
  for (int k = 0; k < KCAT; k += 4) {
    v2f a = *(const v2f*)(arow + k);
    v2f bfrag = *(const v2f*)(wrow + k);
    acc = __builtin_amdgcn_wmma_f32_16x16x4_f32(false, a, false, bfrag,
                                                (short)0, acc, false, false);
  }
  float bv = bias[n0 + m];
  int mbase = (lane >> 4) * 8;  /* D: lanes 0-15 rows r, lanes 16-31 rows r+8 */
  float* outp = logits + n0 + m;
#pragma unroll
  for (int r = 0; r < 8; ++r)
    outp[(size_t)(r + mbase) * VV] = acc[r] + bv;
}

/* ---------------- 11: vocab softmax + assemble final[:, :VEXT] ---------------- */
__global__ __launch_bounds__(256) void k_vocab_softmax(const float* __restrict__ logits,
                                                       const float* __restrict__ pgen,
                                                       float* __restrict__ final_out) {
  int b = blockIdx.x, t = threadIdx.x;
  __shared__ float red[256];
  __shared__ float smx, ssum;
  const float* lr = logits + (size_t)b * VV;
  float mx = -1e30f;
  for (int i = t; i < VV; i += 256) mx = fmaxf(mx, lr[i]);
  red[t] = mx; __syncthreads();
  for (int off = 128; off > 0; off >>= 1) {
    if (t < off) red[t] = fmaxf(red[t], red[t + off]);
    __syncthreads();
  }
  if (t == 0) smx = red[0];
  __syncthreads();
  mx = smx;
  float s = 0.f;
  for (int i = t; i < VV; i += 256) s += __expf(lr[i] - mx);
  red[t] = s; __syncthreads();
  for (int off = 128; off > 0; off >>= 1) {
    if (t < off) red[t] += red[t + off];
    __syncthreads();
  }
  if (t == 0) ssum = red[0];
  __syncthreads();
  float inv = pgen[b] / ssum;
  float* fo = final_out + (size_t)b * VEXT;
  for (int i = t; i < VEXT; i += 256)
    fo[i] = (i < VV) ? __expf(lr[i] - mx) * inv : 0.f;
}

/* ---------------- 12: OOV scatter-add (one thread owns one batch row: deterministic) -- */
__global__ __launch_bounds__(32) void k_scatter(const float* __restrict__ attn_out,
                                                const int* __restrict__ enc_input_oov,
                                                const int* __restrict__ gidx,
                                                const float* __restrict__ pgen,
                                                float* __restrict__ final_out) {
  int b = threadIdx.x;
  if (b >= BB) return;
  float g = 1.f - pgen[b];
  float* fo = final_out + (size_t)b * VEXT;
  for (int i = 0; i < FSZ * WORDLN; ++i) {
    int fsec = i >> 9, w = i & (WORDLN - 1);
    int s = gidx[b * FSZ + fsec] * WORDLN + w;
    float a = attn_out[b * SSN + s];
    if (a != 0.f) fo[enc_input_oov[b * SSN + s]] += g * a;
  }
}

extern "C" void kernel_launch(void* const* d_in, const int* in_sizes, int n_in,
                              void* d_out, int out_size, void* d_ws, size_t ws_size,
                              hipStream_t stream) {
  const int*   inp           = (const int*)  d_in[0];
  const float* h             = (const float*)d_in[1];
  const float* c             = (const float*)d_in[2];
  const float* enc_outputs   = (const float*)d_in[3];
  const float* enc_feature   = (const float*)d_in[4];
  const float* enc_sec       = (const float*)d_in[5];
  const float* enc_mask      = (const float*)d_in[6];
  const float* sec_mask      = (const float*)d_in[7];
  const float* prev_context  = (const float*)d_in[8];
  /* d_in[9] zeros_oov: semantically zeros, handled directly */
  const int*   enc_input_oov = (const int*)  d_in[10];
  const float* coverage      = (const float*)d_in[11];
  const float* focus         = (const float*)d_in[12];
  const float* embedding     = (const float*)d_in[13];
  const float* W_comb        = (const float*)d_in[14];
  const float* b_comb        = (const float*)d_in[15];
  const float* W_ih          = (const float*)d_in[16];
  const float* W_hh          = (const float*)d_in[17];
  const float* b_ih          = (const float*)d_in[18];
  const float* b_hh          = (const float*)d_in[19];
  const float* W_d           = (const float*)d_in[20];
  const float* b_d           = (const float*)d_in[21];
  const float* W_feat        = (const float*)d_in[22];
  const float* v_sec         = (const float*)d_in[23];
  const float* W_dec         = (const float*)d_in[24];
  const float* b_dec         = (const float*)d_in[25];
  const float* w_cov         = (const float*)d_in[26];
  const float* v_att         = (const float*)d_in[27];
  const float* W_pgen        = (const float*)d_in[28];
  const float* b_pgen        = (const float*)d_in[29];
  const float* W_out         = (const float*)d_in[30];
  const float* b_out         = (const float*)d_in[31];

  float* out = (float*)d_out;
  float* o_final = out + OUT_FINAL;
  float* o_hnew  = out + OUT_HNEW;
  float* o_cnew  = out + OUT_CNEW;
  float* o_ctx   = out + OUT_CTX;
  float* o_attn  = out + OUT_ATTN;
  float* o_cov   = out + OUT_COV;
  float* o_focus = out + OUT_FOCUS;
  float* o_sad   = out + OUT_SAD;

  float* ws       = (float*)d_ws;
  float* ws_x     = ws + 0;      /* 16*128   = 2048  */
  float* ws_dec   = ws + 2048;   /* 16*512   = 8192  */
  float* ws_sc    = ws + 10240;  /* 16*8     = 128   */
  float* ws_topk  = ws + 10368;  /* 16*2     = 32    */
  int*   ws_gidx  = (int*)(ws + 10400); /* 16*2 ints */
  float* ws_dfeat = ws + 10432;  /* 16*512   = 8192  */
  float* ws_score = ws + 18624;  /* 16*1024  = 16384 */
  float* ws_cat   = ws + 35008;  /* 16*768   = 12288 */
  float* ws_pgen  = ws + 47296;  /* 16               */
  float* ws_logit = ws + 47312;  /* 16*50000 = 800000 */

  k_x<<<BB, EE, 0, stream>>>(prev_context, inp, embedding, W_comb, b_comb, ws_x);
  k_lstm<<<BB, HH, 0, stream>>>(ws_x, h, c, W_ih, W_hh, b_ih, b_hh,
                                o_hnew, o_cnew, ws_dec, ws_cat);
  k_sec<<<BB * SECLN, 256, 0, stream>>>(enc_sec, W_feat, ws_dec, W_d, b_d, v_sec, ws_sc);
  k_focus<<<1, 32, 0, stream>>>(ws_sc, sec_mask, focus, inp, o_sad, o_focus,
                                ws_topk, ws_gidx);
  k_dfeat<<<BB * 2, 256, 0, stream>>>(ws_dec, W_dec, b_dec, ws_dfeat);
  k_wscore<<<(BB * FSZ * WORDLN) / 8, 256, 0, stream>>>(enc_feature, ws_dfeat, coverage,
                                                        w_cov, v_att, ws_gidx, ws_score);
  k_wsoftmax<<<BB, 256, 0, stream>>>(ws_score, enc_mask, coverage, ws_topk, ws_gidx,
                                     o_attn, o_cov);
  k_context<<<BB, D2H, 0, stream>>>(o_attn, enc_outputs, ws_gidx, o_ctx, ws_cat);
  k_pgen<<<1, BB * 32, 0, stream>>>(o_ctx, ws_dec, ws_x, W_pgen, b_pgen, ws_pgen);
  k_gemm_wmma<<<(VV / 16 + 7) / 8, 256, 0, stream>>>(ws_cat, W_out, b_out, ws_logit);
  k_vocab_softmax<<<BB, 256, 0, stream>>>(ws_logit, ws_pgen, o_final);
  k_scatter<<<1, 32, 0, stream>>>(o_attn, enc_input_oov, ws_gidx, ws_pgen, o_final);
}